// second_derivatives_47940424958271
// MI455X (gfx1250) — compile-verified
//
#include <hip/hip_runtime.h>

// ---------------------------------------------------------------------------
// PINN Hessian (d2u/dxx, d2u/dxy, d2u/dyy) of a 2->128->128->128->1 tanh MLP,
// computed by second-order forward propagation (6 jet channels) with the two
// 128x128 hidden layers done as fp32 WMMA GEMMs (v_wmma_f32_16x16x4_f32).
// ---------------------------------------------------------------------------

constexpr int H       = 128;  // hidden width
constexpr int PTS     = 32;   // collocation points per workgroup
constexpr int THREADS = 256;  // 8 wave32 waves
constexpr int STRIDE  = 132;  // LDS row stride (floats); 132 % 64 == 4 makes the
                              // A-fragment ds_load_b64 pattern bank-conflict-free

typedef __attribute__((ext_vector_type(2))) float v2f;
typedef __attribute__((ext_vector_type(8))) float v8f;

__device__ __forceinline__ v8f wmma4(v2f a, v2f b, v8f c) {
  // D[16x16] = A[16x4] * B[4x16] + C   (fp32, wave32)
  return __builtin_amdgcn_wmma_f32_16x16x4_f32(
      /*neg_a=*/false, a, /*neg_b=*/false, b,
      /*c_mod=*/(short)0, c, /*reuse_a=*/false, /*reuse_b=*/false);
}

// Branch-free tanh: no EXEC-mask manipulation (keeps the WMMA regions clean
// and avoids the divergent OCML path). Odd polynomial for |x| < 0.25 (full
// relative accuracy near zero), exp-based (1-e)/(1+e) elsewhere (saturates
// exactly for large |x|); blended with v_cndmask, sign via copysign.
__device__ __forceinline__ float tanh_bf(float x) {
  const float a  = __builtin_fabsf(x);
  const float e  = __expf(-2.0f * a);                 // v_exp_f32 path
  const float te = (1.0f - e) / (1.0f + e);
  const float a2 = a * a;
  // tanh a = a - a^3/3 + 2a^5/15 - 17a^7/315 + O(a^9)
  const float tp = a * fmaf(a2, fmaf(a2, fmaf(a2, -0.05396825397f,
                                              0.13333333333f),
                                     -0.33333333333f), 1.0f);
  const float t = (a < 0.25f) ? tp : te;
  return __builtin_copysignf(t, x);
}

// tanh chain rule applied elementwise on the 6 jet channels (C-layout regs):
//   h    = tanh(z)
//   h_x  = t' z_x                     h_y  = t' z_y
//   h_ab = t'' z_a z_b + t' z_ab      with t' = 1-t^2, t'' = -2 t (1-t^2)
__device__ __forceinline__ void tanh_chain(const v8f z[6], v8f h[6]) {
#pragma unroll
  for (int r = 0; r < 8; ++r) {
    float zv  = z[0][r], zx = z[1][r], zy = z[2][r];
    float zxx = z[3][r], zxy = z[4][r], zyy = z[5][r];
    float t  = tanh_bf(zv);
    float s  = 1.0f - t * t;   // tanh'
    float sp = -2.0f * t * s;  // tanh''
    h[0][r] = t;
    h[1][r] = s * zx;
    h[2][r] = s * zy;
    h[3][r] = fmaf(sp, zx * zx, s * zxx);
    h[4][r] = fmaf(sp, zx * zy, s * zxy);
    h[5][r] = fmaf(sp, zy * zy, s * zyy);
  }
}

// One hidden layer for the wave's (16-point group g) x (tiles 2q,2q+1) block.
// Reads the 6-channel activation tensor from LDS, multiplies by W (global,
// L2/WGP$-resident), adds bias to the value channel, applies the tanh chain.
// STORE=true : write result channels to the LDS output buffer.
// STORE=false: fold second-derivative channels into W3 dot-product partials.
template <bool STORE>
__device__ __forceinline__ void layer(const float* lin, float* lout,
                                      const float* __restrict__ W,
                                      const float* __restrict__ bias,
                                      const float* __restrict__ W3,
                                      float pdxx[8], float pdxy[8], float pdyy[8],
                                      int g, int q, int l16, int hf) {
  const int prow = g * 16 + l16;               // point row this lane supplies to A
  const float* abase = lin + prow * STRIDE;    // + ch*PTS*STRIDE + k

#pragma unroll
  for (int ti = 0; ti < 2; ++ti) {
    const int tt  = q * 2 + ti;                // global 16-neuron output tile
    const int col = tt * 16 + l16;             // this lane's output neuron (B/C col)

    v8f acc[6];
    const float bv = bias[col];                // bias only enters the value channel
#pragma unroll
    for (int r = 0; r < 8; ++r) {
      acc[0][r] = bv;
      acc[1][r] = 0.0f; acc[2][r] = 0.0f;
      acc[3][r] = 0.0f; acc[4][r] = 0.0f; acc[5][r] = 0.0f;
    }

#pragma unroll 4
    for (int kc = 0; kc < 32; ++kc) {          // K = 128 in chunks of 4
      const int k0 = kc * 4 + hf * 2;          // lanes 16-31 hold K rows +2,+3
      v2f bfrag;                               // B[4x16] fragment of W
      bfrag.x = W[k0 * H + col];
      bfrag.y = W[(k0 + 1) * H + col];
#pragma unroll
      for (int ch = 0; ch < 6; ++ch) {         // 6 jet channels share B
        const v2f afrag = *(const v2f*)(abase + ch * (PTS * STRIDE) + k0);
        acc[ch] = wmma4(afrag, bfrag, acc[ch]);
      }
    }

    v8f h[6];
    tanh_chain(acc, h);

    if constexpr (STORE) {
#pragma unroll
      for (int ch = 0; ch < 6; ++ch) {
#pragma unroll
        for (int r = 0; r < 8; ++r) {
          const int p = g * 16 + r + 8 * hf;   // C-layout: VGPR r, halves = M, M+8
          lout[(ch * PTS + p) * STRIDE + col] = h[ch][r];
        }
      }
    } else {
      const float w3v = W3[col];               // final linear layer: u_ab = a_ab . W3
#pragma unroll
      for (int r = 0; r < 8; ++r) {
        pdxx[r] = fmaf(h[3][r], w3v, pdxx[r]);
        pdxy[r] = fmaf(h[4][r], w3v, pdxy[r]);
        pdyy[r] = fmaf(h[5][r], w3v, pdyy[r]);
      }
    }
  }
}

__global__ void __launch_bounds__(THREADS, 1)
pinn_hess_kernel(const float* __restrict__ X,
                 const float* __restrict__ W0, const float* __restrict__ b0,
                 const float* __restrict__ W1, const float* __restrict__ b1,
                 const float* __restrict__ W2, const float* __restrict__ b2,
                 const float* __restrict__ W3,
                 float* __restrict__ out, int N) {
  __shared__ float buf0[6 * PTS * STRIDE];   // layer-0 output / layer-1 input
  __shared__ float buf1[6 * PTS * STRIDE];   // layer-1 output / layer-2 input
  __shared__ float osum[3 * PTS];            // per-block Hessian accumulators

  const int tid   = threadIdx.x;
  const int lane  = tid & 31;
  const int wave  = tid >> 5;
  const int g     = wave >> 2;   // point group: points [16g, 16g+16)
  const int q     = wave & 3;    // tile pair:   tiles {2q, 2q+1}
  const int l16   = lane & 15;
  const int hf    = lane >> 4;
  const int pbase = (int)blockIdx.x * PTS;

  if (tid < 3 * PTS) osum[tid] = 0.0f;

  // Warm WGP$/L2 with both hidden weight matrices while layer 0 runs.
#pragma unroll
  for (int i = 0; i < 4; ++i) {
    __builtin_prefetch(W1 + ((size_t)(i * THREADS + tid)) * 16, 0, 3);
    __builtin_prefetch(W2 + ((size_t)(i * THREADS + tid)) * 16, 0, 3);
  }

  // ---- Layer 0 (2 -> 128): elementwise jet seeding -----------------------
  // z = x*W0[0,k] + y*W0[1,k] + b0[k];  seeds: z_x=W0[0,k], z_y=W0[1,k], z_ab=0
  for (int idx = tid; idx < PTS * H; idx += THREADS) {
    const int p  = idx >> 7;
    const int k  = idx & (H - 1);
    const int pg = pbase + p;
    float x = 0.0f, y = 0.0f;
    if (pg < N) { x = X[2 * pg]; y = X[2 * pg + 1]; }
    const float wx = W0[k];        // W0[0][k]
    const float wy = W0[H + k];    // W0[1][k]
    const float z  = fmaf(x, wx, fmaf(y, wy, b0[k]));
    const float t  = tanh_bf(z);
    const float s  = 1.0f - t * t;
    const float sp = -2.0f * t * s;
    float* dst = buf0 + p * STRIDE + k;
    dst[0 * PTS * STRIDE] = t;
    dst[1 * PTS * STRIDE] = s * wx;
    dst[2 * PTS * STRIDE] = s * wy;
    dst[3 * PTS * STRIDE] = sp * wx * wx;
    dst[4 * PTS * STRIDE] = sp * wx * wy;
    dst[5 * PTS * STRIDE] = sp * wy * wy;
  }
  __syncthreads();

  float pdxx[8], pdxy[8], pdyy[8];
#pragma unroll
  for (int r = 0; r < 8; ++r) { pdxx[r] = 0.0f; pdxy[r] = 0.0f; pdyy[r] = 0.0f; }

  // ---- Layer 1 (GEMM + tanh chain), buf0 -> buf1 -------------------------
  layer<true>(buf0, buf1, W1, b1, nullptr, pdxx, pdxy, pdyy, g, q, l16, hf);
  __syncthreads();

  // ---- Layer 2 (GEMM + tanh chain) fused with W3 dot products ------------
  layer<false>(buf1, nullptr, W2, b2, W3, pdxx, pdxy, pdyy, g, q, l16, hf);

  // Butterfly-reduce the 16 neuron-columns held by each lane half.
#pragma unroll
  for (int r = 0; r < 8; ++r) {
#pragma unroll
    for (int m = 1; m < 16; m <<= 1) {
      pdxx[r] += __shfl_xor(pdxx[r], m, 32);
      pdxy[r] += __shfl_xor(pdxy[r], m, 32);
      pdyy[r] += __shfl_xor(pdyy[r], m, 32);
    }
  }
  if (l16 == 0) {  // lanes 0 and 16 carry the per-point totals for this wave's k-range
    const int p0 = g * 16 + 8 * hf;
#pragma unroll
    for (int r = 0; r < 8; ++r) {
      atomicAdd(&osum[0 * PTS + p0 + r], pdxx[r]);  // ds_add_f32
      atomicAdd(&osum[1 * PTS + p0 + r], pdxy[r]);
      atomicAdd(&osum[2 * PTS + p0 + r], pdyy[r]);
    }
  }
  __syncthreads();

  // Output layout: [du_dxx (N) | du_dxy (N) | du_dyy (N)]
  if (tid < 3 * PTS) {
    const int c  = tid >> 5;
    const int p  = tid & 31;
    const int pg = pbase + p;
    if (pg < N) out[(size_t)c * N + pg] = osum[tid];
  }
}

extern "C" void kernel_launch(void* const* d_in, const int* in_sizes, int n_in,
                              void* d_out, int out_size, void* d_ws, size_t ws_size,
                              hipStream_t stream) {
  const float* X  = (const float*)d_in[0];
  const float* W0 = (const float*)d_in[1];
  const float* b0 = (const float*)d_in[2];
  const float* W1 = (const float*)d_in[3];
  const float* b1 = (const float*)d_in[4];
  const float* W2 = (const float*)d_in[5];
  const float* b2 = (const float*)d_in[6];
  const float* W3 = (const float*)d_in[7];
  // d_in[8] = b3: constant bias, zero second derivative -> unused.
  float* out = (float*)d_out;

  const int N      = in_sizes[0] / 2;
  const int blocks = (N + PTS - 1) / PTS;
  hipLaunchKernelGGL(pinn_hess_kernel, dim3(blocks), dim3(THREADS), 0, stream,
                     X, W0, b0, W1, b1, W2, b2, W3, out, N);
}